// MultiBoxLoss_1417339207999
// MI455X (gfx1250) — compile-verified
//
#include <hip/hip_runtime.h>
#include <math.h>

// ---------------- Problem constants ----------------
#define C_CLS 21
#define RPB   256          // rows (b,p pairs) per block in pass 1
#define NEGPOS_RATIO 3

typedef __attribute__((ext_vector_type(2))) float v2f;
typedef __attribute__((ext_vector_type(8))) float v8f;
typedef __attribute__((ext_vector_type(4))) int   v4i;

typedef __attribute__((address_space(1))) v4i* as1_v4ip;
typedef __attribute__((address_space(3))) v4i* as3_v4ip;

// Monotonic float <-> uint mapping (ascending order preserved)
__device__ __forceinline__ unsigned mapf(float f) {
  unsigned u = __float_as_uint(f);
  return (u & 0x80000000u) ? ~u : (u | 0x80000000u);
}
__device__ __forceinline__ float unmapf(unsigned u) {
  unsigned b = (u & 0x80000000u) ? (u & 0x7FFFFFFFu) : ~u;
  return __uint_as_float(b);
}

// ---------------------------------------------------------------------------
// Pass 1: per (b,p) row — logsumexp over C=21 classes (staged via async LDS),
// loss_c for mining, positive CE + smooth-L1 partial sums, num_pos per batch.
// ---------------------------------------------------------------------------
__global__ void pass1_kernel(const float* __restrict__ loc,
                             const float* __restrict__ conf,
                             const float* __restrict__ loc_t,
                             const int*   __restrict__ conf_t,
                             float* __restrict__ lossc,     // [rows]
                             float* __restrict__ partials,  // [gridDim.x]
                             int*   __restrict__ numpos,    // [B]
                             int P, int rows, int Bn) {
  __shared__ __attribute__((aligned(16))) float sconf[RPB * C_CLS];
  __shared__ float sred[RPB];
  __shared__ int   scnt[2];

  const int tid = threadIdx.x;
  const int r0  = blockIdx.x * RPB;
  const int vr  = min(RPB, rows - r0);
  const int b_first = r0 / P;

  if (tid < 2) scnt[tid] = 0;

  // ---- stage this block's conf tile (vr rows x 21 f32) into LDS ----
  const int nbytes  = vr * C_CLS * 4;
  const int nchunks = (nbytes + 15) >> 4;          // 16B chunks, base 16B-aligned
  const char* gsrc  = (const char*)conf + (size_t)r0 * C_CLS * 4;

#if __has_builtin(__builtin_amdgcn_global_load_async_to_lds_b128) && \
    __has_builtin(__builtin_amdgcn_s_wait_asynccnt)
  for (int i = tid; i < nchunks; i += blockDim.x) {
    __builtin_amdgcn_global_load_async_to_lds_b128(
        (as1_v4ip)(gsrc + (size_t)i * 16),
        (as3_v4ip)((char*)sconf + (size_t)i * 16),
        0, 0);
  }
  __builtin_amdgcn_s_wait_asynccnt(0);
#else
  for (int i = tid; i < nchunks; i += blockDim.x) {
    ((float4*)sconf)[i] = ((const float4*)gsrc)[i];
  }
#endif
  __syncthreads();

  float acc = 0.0f;  // merged: S_pos (positive CE) + smooth-L1 contribution
  if (tid < vr) {
    const int r = r0 + tid;
    const float* row = &sconf[tid * C_CLS];

    float m = row[0];
#pragma unroll
    for (int c = 1; c < C_CLS; ++c) m = fmaxf(m, row[c]);
    float s = 0.0f;
#pragma unroll
    for (int c = 0; c < C_CLS; ++c) s += __expf(row[c] - m);
    const float lse = m + __logf(s);

    const int  t   = conf_t[r];
    const float ce = lse - row[t];
    const bool pos = (t > 0);
    lossc[r] = pos ? 0.0f : ce;     // positives excluded from mining

    if (pos) {
      atomicAdd(&scnt[(r / P) - b_first], 1);
      acc = ce;                      // CE of positives always counted
      const float4 l4 = ((const float4*)loc)[r];
      const float4 t4 = ((const float4*)loc_t)[r];
      float dx[4] = {l4.x - t4.x, l4.y - t4.y, l4.z - t4.z, l4.w - t4.w};
#pragma unroll
      for (int i = 0; i < 4; ++i) {
        const float ad = fabsf(dx[i]);
        acc += (ad < 1.0f) ? 0.5f * dx[i] * dx[i] : ad - 0.5f;
      }
    }
  }

  // deterministic block tree reduction
  sred[tid] = acc;
  __syncthreads();
  for (int off = blockDim.x >> 1; off > 0; off >>= 1) {
    if (tid < off) sred[tid] += sred[tid + off];
    __syncthreads();
  }
  if (tid == 0) partials[blockIdx.x] = sred[0];

  if (tid < 2) {
    const int b = b_first + tid;
    if (scnt[tid] > 0 && b < Bn) atomicAdd(&numpos[b], scnt[tid]);
  }
}

// ---------------------------------------------------------------------------
// Pass 2: one block per batch row — radix-select the k-th largest loss_c and
// sum the top-k values (tie-safe: equal values contribute equally).
// ---------------------------------------------------------------------------
__global__ void select_kernel(const float* __restrict__ lossc,
                              const int*   __restrict__ numpos,
                              float* __restrict__ rowsum,   // [B]
                              int P) {
  __shared__ int      hist[256];
  __shared__ float    fred[1024];
  __shared__ unsigned sh_prefix;
  __shared__ int      sh_rem;

  const int b   = blockIdx.x;
  const int tid = threadIdx.x;
  const int nt  = blockDim.x;
  const float* lc = lossc + (size_t)b * P;

  const int np = numpos[b];
  const long long kll = (long long)NEGPOS_RATIO * (long long)np;
  const int kk = (kll > (long long)(P - 1)) ? (P - 1) : (int)kll;
  if (kk <= 0) { if (tid == 0) rowsum[b] = 0.0f; return; }

  unsigned prefix = 0;
  int remaining = kk;
  for (int pass = 0; pass < 4; ++pass) {
    for (int i = tid; i < 256; i += nt) hist[i] = 0;
    __syncthreads();
    const int shift = 24 - 8 * pass;
    for (int i = tid; i < P; i += nt) {
      const unsigned key = mapf(lc[i]);
      if (pass == 0 || (key >> (shift + 8)) == prefix)
        atomicAdd(&hist[(key >> shift) & 0xFF], 1);
    }
    __syncthreads();
    if (tid == 0) {
      int c = 0, chosen = 0;
      for (int bin = 255; bin >= 0; --bin) {
        const int h = hist[bin];
        if (c + h >= remaining) { chosen = bin; break; }
        c += h;
      }
      sh_prefix = (prefix << 8) | (unsigned)chosen;
      sh_rem = remaining - c;   // elements equal to the threshold to include
    }
    __syncthreads();
    prefix = sh_prefix;
    remaining = sh_rem;
    __syncthreads();
  }

  const unsigned T = prefix;       // exact key of the k-th largest value
  const float tv = unmapf(T);
  float sg = 0.0f;
  for (int i = tid; i < P; i += nt) {
    const float v = lc[i];
    if (mapf(v) > T) sg += v;      // strictly-greater values
  }
  fred[tid] = sg;
  __syncthreads();
  for (int off = nt >> 1; off > 0; off >>= 1) {
    if (tid < off) fred[tid] += fred[tid + off];
    __syncthreads();
  }
  if (tid == 0) rowsum[b] = fred[0] + (float)remaining * tv;
}

// ---------------------------------------------------------------------------
// Final: one wave sums all partials + per-row sums. Cross-lane fold done with
// V_WMMA_F32_16X16X4_F32 against an all-ones B (exact: x*1.0, f32 accumulate).
//   A[m,0]=acc(lane m), A[m,2]=acc(lane m+16) -> D[m,0] = acc[m]+acc[m+16]
//   lane 0 holds rows 0..7 of col 0, lane 16 holds rows 8..15 of col 0.
// ---------------------------------------------------------------------------
__global__ void final_kernel(const float* __restrict__ partials, int nPart,
                             const float* __restrict__ rowsum,   int nRow,
                             float* __restrict__ out) {
  const int lane = threadIdx.x;          // 32 threads, 1 wave, EXEC all ones
  float acc = 0.0f;
  for (int i = lane; i < nPart; i += 32) acc += partials[i];
  for (int i = lane; i < nRow;  i += 32) acc += rowsum[i];

#if __has_builtin(__builtin_amdgcn_wmma_f32_16x16x4_f32)
  v2f a;  a[0] = acc;  a[1] = 0.0f;
  v2f bo; bo[0] = 1.0f; bo[1] = 1.0f;
  v8f c = {};
  c = __builtin_amdgcn_wmma_f32_16x16x4_f32(false, a, false, bo,
                                            (short)0, c, false, false);
  const float s = c[0] + c[1] + c[2] + c[3] + c[4] + c[5] + c[6] + c[7];
  const float o = __shfl(s, 16, 32);     // lane16's half (rows 8..15)
  if (lane == 0) out[0] = s + o;
#else
  for (int off = 16; off > 0; off >>= 1) acc += __shfl_down(acc, off, 32);
  if (lane == 0) out[0] = acc;
#endif
}

__global__ void init_kernel(int* __restrict__ numpos, float* __restrict__ out,
                            int Bn) {
  const int i = threadIdx.x + blockIdx.x * blockDim.x;
  if (i < Bn) numpos[i] = 0;
  if (i == 0) out[0] = 0.0f;
}

// ---------------------------------------------------------------------------
extern "C" void kernel_launch(void* const* d_in, const int* in_sizes, int n_in,
                              void* d_out, int out_size, void* d_ws, size_t ws_size,
                              hipStream_t stream) {
  (void)n_in; (void)out_size; (void)ws_size;
  const float* loc    = (const float*)d_in[0];
  const float* conf   = (const float*)d_in[1];
  const float* priors = (const float*)d_in[2]; (void)priors;  // unused by ref
  const float* loc_t  = (const float*)d_in[3];
  const int*   conf_t = (const int*)d_in[4];

  const int BP = in_sizes[4];          // B * P
  const int P  = in_sizes[2] / 4;      // priors is [P,4]
  const int Bn = BP / P;

  const int nBlocks = (BP + RPB - 1) / RPB;

  char* ws = (char*)d_ws;
  float* lossc = (float*)ws;
  size_t off = ((size_t)BP * 4 + 255) & ~(size_t)255;
  float* partials = (float*)(ws + off);
  off = (off + (size_t)nBlocks * 4 + 255) & ~(size_t)255;
  float* rowsum = (float*)(ws + off);
  off = (off + (size_t)Bn * 4 + 255) & ~(size_t)255;
  int* numpos = (int*)(ws + off);

  init_kernel<<<1, 256, 0, stream>>>(numpos, (float*)d_out, Bn);
  pass1_kernel<<<nBlocks, RPB, 0, stream>>>(loc, conf, loc_t, conf_t,
                                            lossc, partials, numpos,
                                            P, BP, Bn);
  select_kernel<<<Bn, 1024, 0, stream>>>(lossc, numpos, rowsum, P);
  final_kernel<<<1, 32, 0, stream>>>(partials, nBlocks, rowsum, Bn,
                                     (float*)d_out);
}